// _SimpleSegmentationModel_mltfrm_82987358093347
// MI455X (gfx1250) — compile-verified
//
#include <hip/hip_runtime.h>
#include <hip/hip_bf16.h>
#include <math.h>

typedef __attribute__((ext_vector_type(2))) float v2f;
typedef __attribute__((ext_vector_type(8))) float v8f;

#define C_DIM 2048
#define L_DIM 1024
#define NCLS 21
#define LDST 136            // padded LDS row stride (floats): half-wave K-offset reads bank-disjoint
#define SLAB (16 * LDST)    // floats per 16xK slab

// ---- CDNA5 async global->LDS copy (ASYNCcnt path), 16B per lane ----
__device__ __forceinline__ void async_ld16(unsigned lds_addr, const float* g) {
    asm volatile("global_load_async_to_lds_b128 %0, %1, off"
                 :: "v"(lds_addr), "v"(g) : "memory");
}
__device__ __forceinline__ void wait_async0() {
    asm volatile("s_wait_asynccnt 0x0" ::: "memory");
}

// ---------------- Weff = w_cls [21,2048] @ w_red [2048,8192] -> [21,8192] ----------------
__global__ __launch_bounds__(256) void weff_kernel(const float* __restrict__ w_cls,
                                                   const float* __restrict__ w_red,
                                                   float* __restrict__ Weff) {
    int c = blockIdx.x * 256 + threadIdx.x;      // 0..8191
    float acc[NCLS];
#pragma unroll
    for (int k = 0; k < NCLS; ++k) acc[k] = 0.f;
    for (int o = 0; o < C_DIM; ++o) {
        float w = w_red[(size_t)o * 8192 + c];   // coalesced; w_cls reads are wave-uniform
#pragma unroll
        for (int k = 0; k < NCLS; ++k) acc[k] += w_cls[k * C_DIM + o] * w;
    }
#pragma unroll
    for (int k = 0; k < NCLS; ++k) Weff[(size_t)k * 8192 + c] = acc[k];
}

// ---------------- z[zidx][b] = Weff_segment @ xf[b]  ([21,2048]x[2048,1024]) ----------------
__global__ __launch_bounds__(256) void z_kernel(const float* __restrict__ xf,
                                                const float* __restrict__ Weff,
                                                float* __restrict__ z) {
    int b = blockIdx.z;                          // 0..3
    int zidx = blockIdx.y;                       // 0: seg0(xf)  1: seg1(a0)  2: seg2+seg3(a1)
    int j = blockIdx.x * 256 + threadIdx.x;      // 0..1023
    const float* X = xf + (size_t)b * C_DIM * L_DIM;
    float acc[NCLS];
#pragma unroll
    for (int k = 0; k < NCLS; ++k) acc[k] = 0.f;
    for (int c = 0; c < C_DIM; ++c) {
        float v = X[(size_t)c * L_DIM + j];
        if (zidx == 0) {
#pragma unroll
            for (int k = 0; k < NCLS; ++k) acc[k] += Weff[(size_t)k * 8192 + c] * v;
        } else if (zidx == 1) {
#pragma unroll
            for (int k = 0; k < NCLS; ++k) acc[k] += Weff[(size_t)k * 8192 + 2048 + c] * v;
        } else {
#pragma unroll
            for (int k = 0; k < NCLS; ++k)
                acc[k] += (Weff[(size_t)k * 8192 + 4096 + c] + Weff[(size_t)k * 8192 + 6144 + c]) * v;
        }
    }
#pragma unroll
    for (int k = 0; k < NCLS; ++k)
        z[(size_t)((zidx * 4 + b) * NCLS + k) * L_DIM + j] = acc[k];
}

// ---------------- dot[bb] = xf[b]^T @ other[b]  (fp32 WMMA 16x16x4) ----------------
// bb = branch*4 + batch; branch0 -> x0, branch1 -> x1.  128x128 tile / block (8 waves).
// K slabs staged global->LDS via async copies (ASYNCcnt), double-buffered: loads for
// slab K+1 fly while the 32 WMMAs of slab K execute; one barrier per slab.
__global__ __launch_bounds__(256) void dot_kernel(const float* __restrict__ xf,
                                                  const float* __restrict__ x0,
                                                  const float* __restrict__ x1,
                                                  float* __restrict__ dot) {
    __shared__ float smem[4 * SLAB];             // [buf0: A,B][buf1: A,B] ~ 34.8 KB
    const int bb = blockIdx.z;
    const int batch = bb & 3, branch = bb >> 2;
    const float* A  = xf + (size_t)batch * C_DIM * L_DIM;
    const float* Bm = (branch == 0 ? x0 : x1) + (size_t)batch * C_DIM * L_DIM;
    const int l0 = blockIdx.y * 128;
    const int j0 = blockIdx.x * 128;
    const int tid = threadIdx.x;
    const int lane = tid & 31;
    const int wave = tid >> 5;
    const int wm = wave & 1;     // 2 waves over M (64 rows each)
    const int wn = wave >> 1;    // 4 waves over N (32 cols each)
    const int mlane = lane & 15;
    const int koff = (lane < 16) ? 0 : 2;      // A/B fragment half-wave K offset (ISA 7.12.2)
    const int mbase = (lane < 16) ? 0 : 8;     // C/D fragment half-wave M offset

    // staging task for this thread: two 16B chunks per matrix per slab
    const int q0 = tid, q1 = tid + 256;        // float4 slot ids (512 per 16x128 slab)
    const int row0 = q0 >> 5, c40 = (q0 & 31) * 4;
    const int row1 = q1 >> 5, c41 = (q1 & 31) * 4;

    v8f acc[4][2] = {};

    // prologue: slab 0 into buffer 0
    {
        float* AsB = smem;
        float* BsB = smem + SLAB;
        async_ld16((unsigned)(uintptr_t)(AsB + row0 * LDST + c40), A  + (size_t)row0 * L_DIM + l0 + c40);
        async_ld16((unsigned)(uintptr_t)(AsB + row1 * LDST + c41), A  + (size_t)row1 * L_DIM + l0 + c41);
        async_ld16((unsigned)(uintptr_t)(BsB + row0 * LDST + c40), Bm + (size_t)row0 * L_DIM + j0 + c40);
        async_ld16((unsigned)(uintptr_t)(BsB + row1 * LDST + c41), Bm + (size_t)row1 * L_DIM + j0 + c41);
    }

    for (int kt = 0; kt < C_DIM; kt += 16) {
        const int buf = (kt >> 4) & 1;
        wait_async0();        // this wave's slab-kt loads complete
        __syncthreads();      // all waves' loads complete; all waves done with buf^1
        if (kt + 16 < C_DIM) {
            float* AsB = smem + (buf ^ 1) * 2 * SLAB;
            float* BsB = AsB + SLAB;
            const int kn = kt + 16;
            async_ld16((unsigned)(uintptr_t)(AsB + row0 * LDST + c40), A  + (size_t)(kn + row0) * L_DIM + l0 + c40);
            async_ld16((unsigned)(uintptr_t)(AsB + row1 * LDST + c41), A  + (size_t)(kn + row1) * L_DIM + l0 + c41);
            async_ld16((unsigned)(uintptr_t)(BsB + row0 * LDST + c40), Bm + (size_t)(kn + row0) * L_DIM + j0 + c40);
            async_ld16((unsigned)(uintptr_t)(BsB + row1 * LDST + c41), Bm + (size_t)(kn + row1) * L_DIM + j0 + c41);
        }
        const float* As = smem + buf * 2 * SLAB;
        const float* Bs = As + SLAB;
#pragma unroll
        for (int s = 0; s < 16; s += 4) {
#pragma unroll
            for (int tm = 0; tm < 4; ++tm) {
                int m = wm * 64 + tm * 16 + mlane;
                v2f a;
                a[0] = As[(s + koff) * LDST + m];
                a[1] = As[(s + koff + 1) * LDST + m];
#pragma unroll
                for (int tn = 0; tn < 2; ++tn) {
                    int n = wn * 32 + tn * 16 + mlane;
                    v2f bf;
                    bf[0] = Bs[(s + koff) * LDST + n];
                    bf[1] = Bs[(s + koff + 1) * LDST + n];
                    acc[tm][tn] = __builtin_amdgcn_wmma_f32_16x16x4_f32(
                        false, a, false, bf, (short)0, acc[tm][tn], false, false);
                }
            }
        }
    }

    float* Dp = dot + (size_t)bb * L_DIM * L_DIM;
#pragma unroll
    for (int tm = 0; tm < 4; ++tm)
#pragma unroll
        for (int tn = 0; tn < 2; ++tn)
#pragma unroll
            for (int v = 0; v < 8; ++v) {
                int mm = l0 + wm * 64 + tm * 16 + mbase + v;
                int nn = j0 + wn * 32 + tn * 16 + mlane;
                Dp[(size_t)mm * L_DIM + nn] = acc[tm][tn][v];
            }
}

// ---------------- global softmax reductions (deterministic two-stage) ----------------
__global__ __launch_bounds__(256) void rmax_kernel(const float* __restrict__ dot,
                                                   float* __restrict__ pmax) {
    __shared__ float s[256];
    int bb = blockIdx.y, chunk = blockIdx.x, tid = threadIdx.x;
    const float* p = dot + (size_t)bb * 1048576 + (size_t)chunk * 32768;
    float m = -INFINITY;
    for (int i = tid; i < 32768; i += 256) m = fmaxf(m, p[i]);
    s[tid] = m; __syncthreads();
    for (int st = 128; st > 0; st >>= 1) { if (tid < st) s[tid] = fmaxf(s[tid], s[tid + st]); __syncthreads(); }
    if (tid == 0) pmax[bb * 32 + chunk] = s[0];
}
__global__ void rmax2_kernel(const float* __restrict__ pmax, float* __restrict__ gmax) {
    __shared__ float s[32];
    int bb = blockIdx.x, tid = threadIdx.x;
    s[tid] = pmax[bb * 32 + tid]; __syncthreads();
    for (int st = 16; st > 0; st >>= 1) { if (tid < st) s[tid] = fmaxf(s[tid], s[tid + st]); __syncthreads(); }
    if (tid == 0) gmax[bb] = s[0];
}
__global__ __launch_bounds__(256) void rsum_kernel(const float* __restrict__ dot,
                                                   const float* __restrict__ gmax,
                                                   float* __restrict__ psum) {
    __shared__ float s[256];
    int bb = blockIdx.y, chunk = blockIdx.x, tid = threadIdx.x;
    const float* p = dot + (size_t)bb * 1048576 + (size_t)chunk * 32768;
    float gm = gmax[bb];
    float acc = 0.f;
    for (int i = tid; i < 32768; i += 256) acc += expf(p[i] - gm);
    s[tid] = acc; __syncthreads();
    for (int st = 128; st > 0; st >>= 1) { if (tid < st) s[tid] += s[tid + st]; __syncthreads(); }
    if (tid == 0) psum[bb * 32 + chunk] = s[0];
}
__global__ void rsum2_kernel(const float* __restrict__ psum, float* __restrict__ gsum) {
    __shared__ float s[32];
    int bb = blockIdx.x, tid = threadIdx.x;
    s[tid] = psum[bb * 32 + tid]; __syncthreads();
    for (int st = 16; st > 0; st >>= 1) { if (tid < st) s[tid] += s[tid + st]; __syncthreads(); }
    if (tid == 0) gsum[bb] = s[0];
}

// ---------------- U[bb] = z_branch @ exp(dot[bb] - gmax)  ([21,1024]x[1024,1024]) ----------------
__global__ __launch_bounds__(256) void u_kernel(const float* __restrict__ dot,
                                                const float* __restrict__ z,
                                                const float* __restrict__ gmax,
                                                float* __restrict__ U) {
    int bb = blockIdx.y;
    int j = blockIdx.x * 256 + threadIdx.x;
    int b = bb & 3, branch = bb >> 2;
    int zidx = (branch == 0) ? 1 : 2;
    const float* D = dot + (size_t)bb * 1048576;
    const float* Z = z + (size_t)((zidx * 4 + b) * NCLS) * L_DIM;
    float gm = gmax[bb];
    float acc[NCLS];
#pragma unroll
    for (int k = 0; k < NCLS; ++k) acc[k] = 0.f;
    for (int l = 0; l < L_DIM; ++l) {
        float e = expf(D[(size_t)l * L_DIM + j] - gm);
#pragma unroll
        for (int k = 0; k < NCLS; ++k) acc[k] += Z[(size_t)k * L_DIM + l] * e;
    }
    float* Up = U + (size_t)bb * NCLS * L_DIM;
#pragma unroll
    for (int k = 0; k < NCLS; ++k) Up[(size_t)k * L_DIM + j] = acc[k];
}

// ---------------- logits = zx + U0/sum0 + U1/sum1 + bias ----------------
__global__ void logits_kernel(const float* __restrict__ z, const float* __restrict__ U,
                              const float* __restrict__ gsum, const float* __restrict__ b_cls,
                              float* __restrict__ logits) {
    int idx = blockIdx.x * 256 + threadIdx.x;        // [b][k][j]
    if (idx >= 4 * NCLS * L_DIM) return;
    int j = idx & 1023;
    int k = (idx >> 10) % NCLS;
    int b = idx / (NCLS * L_DIM);
    float v = z[(size_t)((0 * 4 + b) * NCLS + k) * L_DIM + j];               // zx
    v += U[(size_t)(b) * NCLS * L_DIM + (size_t)k * L_DIM + j] / gsum[b];    // branch0 (a0)
    v += U[(size_t)(4 + b) * NCLS * L_DIM + (size_t)k * L_DIM + j] / gsum[4 + b]; // branch1 (a1 x2)
    v += b_cls[k];
    logits[idx] = v;
}

// ---------------- bilinear upsample 32x32 -> 512x512, half-pixel centers ----------------
__global__ void upsample_kernel(const float* __restrict__ logits, float* __restrict__ out) {
    size_t idx = (size_t)blockIdx.x * 256 + threadIdx.x;
    if (idx >= (size_t)4 * NCLS * 512 * 512) return;
    int x = (int)(idx & 511);
    int y = (int)((idx >> 9) & 511);
    int ck = (int)(idx >> 18);                       // b*21+k
    float fy = (y + 0.5f) * 0.0625f - 0.5f;
    float fx = (x + 0.5f) * 0.0625f - 0.5f;
    float y0f = floorf(fy), x0f = floorf(fx);
    float wy = fy - y0f, wx = fx - x0f;
    int y0 = min(max((int)y0f, 0), 31), y1 = min(max((int)y0f + 1, 0), 31);
    int xa = min(max((int)x0f, 0), 31), xb = min(max((int)x0f + 1, 0), 31);
    const float* Lp = logits + (size_t)ck * 1024;
    float v00 = Lp[y0 * 32 + xa], v01 = Lp[y0 * 32 + xb];
    float v10 = Lp[y1 * 32 + xa], v11 = Lp[y1 * 32 + xb];
    out[idx] = (1.f - wy) * ((1.f - wx) * v00 + wx * v01) + wy * ((1.f - wx) * v10 + wx * v11);
}

extern "C" void kernel_launch(void* const* d_in, const int* in_sizes, int n_in,
                              void* d_out, int out_size, void* d_ws, size_t ws_size,
                              hipStream_t stream) {
    const float* xf    = (const float*)d_in[0];
    const float* x0    = (const float*)d_in[1];
    const float* x1    = (const float*)d_in[2];
    // d_in[3] (x2_feat) intentionally unused: reference discards a2.
    const float* w_red = (const float*)d_in[4];
    const float* w_cls = (const float*)d_in[5];
    const float* b_cls = (const float*)d_in[6];
    float* out = (float*)d_out;

    float* ws = (float*)d_ws;
    float* Weff   = ws;                        // 21*8192        = 172032
    float* dotb   = Weff + 172032;             // 8*1024*1024    = 8388608
    float* zbuf   = dotb + 8388608;            // 3*4*21*1024    = 258048
    float* Ubuf   = zbuf + 258048;             // 8*21*1024      = 172032
    float* logitsb= Ubuf + 172032;             // 4*21*1024      = 86016
    float* pmax   = logitsb + 86016;           // 256
    float* gmax   = pmax + 256;                // 8
    float* psum   = gmax + 8;                  // 256
    float* gsum   = psum + 256;                // 8

    weff_kernel<<<32, 256, 0, stream>>>(w_cls, w_red, Weff);
    z_kernel<<<dim3(4, 3, 4), 256, 0, stream>>>(xf, Weff, zbuf);
    dot_kernel<<<dim3(8, 8, 8), 256, 0, stream>>>(xf, x0, x1, dotb);
    rmax_kernel<<<dim3(32, 8), 256, 0, stream>>>(dotb, pmax);
    rmax2_kernel<<<8, 32, 0, stream>>>(pmax, gmax);
    rsum_kernel<<<dim3(32, 8), 256, 0, stream>>>(dotb, gmax, psum);
    rsum2_kernel<<<8, 32, 0, stream>>>(psum, gsum);
    u_kernel<<<dim3(4, 8), 256, 0, stream>>>(dotb, zbuf, gmax, Ubuf);
    logits_kernel<<<(4 * NCLS * L_DIM + 255) / 256, 256, 0, stream>>>(zbuf, Ubuf, gsum, b_cls, logitsb);
    upsample_kernel<<<(int)(((size_t)4 * NCLS * 512 * 512 + 255) / 256), 256, 0, stream>>>(logitsb, out);
}